// VanillaMetric_31112743092674
// MI455X (gfx1250) — compile-verified
//
#include <hip/hip_runtime.h>

// VanillaMetric sparse-adjacency build (rational kernel, symmetric).
// Memory-bound scatter/gather: no matrix ops -> no WMMA (would be dead code).
// CDNA5 paths: wave32 shuffle reduction (ds_bpermute), global_atomic_add_f32
// fire-and-forget L2 atomics, b128 loads + non-temporal b128 streaming stores.

#define TPB 256  // 8 wave32s per block

// native clang vector types (required by __builtin_nontemporal_store and to
// guarantee b128 codegen)
typedef float v4f __attribute__((ext_vector_type(4)));
typedef int   v4i __attribute__((ext_vector_type(4)));

// ---- zero the row/col segment-sum accumulators in workspace -----------------
__global__ __launch_bounds__(TPB) void zero_sums(float* __restrict__ sums, int n) {
  int i = blockIdx.x * TPB + threadIdx.x;
  if (i < n) sums[i] = 0.0f;
}

// rational kernel weight for one edge (vertex gathers are L2-resident: 6 MB)
__device__ __forceinline__ float edge_weight(const float* __restrict__ verts,
                                             int s, int d) {
  float sx = verts[3 * s + 0], sy = verts[3 * s + 1], sz = verts[3 * s + 2];
  float tx = verts[3 * d + 0] - sx;
  float ty = verts[3 * d + 1] - sy;
  float tz = verts[3 * d + 2] - sz;
  float dist2 = tx * tx + ty * ty + tz * tz;
  return 1.0f / (1.0f + dist2);
}

// ---- pass 1: w per edge, scatter-add into row_sum[src] / col_sum[dst] -------
// 4 edges per thread. b128 edge loads. Row sums: edges arrive sorted by src
// (32 per vertex), so a thread's 4 edges share one src and an 8-lane group's
// 32 edges share one src. Detect dynamically; collapse 32 conflicting atomics
// into one group-reduced atomic. Generic per-edge atomic fallback otherwise.
__global__ __launch_bounds__(TPB) void edge_pass1(
    const float* __restrict__ verts,
    const int*   __restrict__ esrc,
    const int*   __restrict__ edst,
    float* __restrict__ row_sum,
    float* __restrict__ col_sum,
    int E) {
  const int i = blockIdx.x * TPB + threadIdx.x;
  const int e = i * 4;
  if (e >= E) return;

  if (((E & 3) == 0) && (e + 3 < E)) {
    // ---- vector path: 4 edges, b128 index loads ----
    v4i s4 = *(const v4i*)(esrc + e);
    v4i d4 = *(const v4i*)(edst + e);

    float w0 = edge_weight(verts, s4.x, d4.x);
    float w1 = edge_weight(verts, s4.y, d4.y);
    float w2 = edge_weight(verts, s4.z, d4.z);
    float w3 = edge_weight(verts, s4.w, d4.w);

    // dst addresses are distinct within a wave for this edge list:
    // conflict-free fire-and-forget global_atomic_add_f32 into L2.
    atomicAdd(&col_sum[d4.x], w0);
    atomicAdd(&col_sum[d4.y], w1);
    atomicAdd(&col_sum[d4.z], w2);
    atomicAdd(&col_sum[d4.w], w3);

    const bool tuni = (s4.x == s4.y) && (s4.x == s4.z) && (s4.x == s4.w);
    const float wsum = (w0 + w1) + (w2 + w3);

    const int lane  = threadIdx.x & 31;      // wave32
    const int gbase = lane & ~7;             // 8-lane group leader
    unsigned long long act = __ballot(1);
    int sfirst = __shfl(s4.x, gbase);
    unsigned long long uni = __ballot(tuni && (s4.x == sfirst));
    bool grp_ok = (act == 0xFFFFFFFFull) &&
                  (((uni >> gbase) & 0xFFull) == 0xFFull);
    if (grp_ok) {
      float t = wsum;                        // butterfly over 8-lane group
      t += __shfl_xor(t, 4);
      t += __shfl_xor(t, 2);
      t += __shfl_xor(t, 1);
      if (lane == gbase) atomicAdd(&row_sum[s4.x], t);
    } else if (tuni) {
      atomicAdd(&row_sum[s4.x], wsum);
    } else {
      atomicAdd(&row_sum[s4.x], w0);
      atomicAdd(&row_sum[s4.y], w1);
      atomicAdd(&row_sum[s4.z], w2);
      atomicAdd(&row_sum[s4.w], w3);
    }
  } else {
    // ---- scalar tail / unaligned fallback ----
    int lim = e + 4 < E ? e + 4 : E;
    for (int k = e; k < lim; ++k) {
      int s = esrc[k], d = edst[k];
      float w = edge_weight(verts, s, d);
      atomicAdd(&col_sum[d], w);
      atomicAdd(&row_sum[s], w);
    }
  }
}

// ---- pass 2: recompute w (cheaper than a 128MB HBM round trip), normalize ---
// Edge list stays L2-resident after pass 1 (128MB edges + 6MB verts + 4MB sums
// < 192MB L2); NT b128 stores keep the 128MB output from evicting it.
__global__ __launch_bounds__(TPB) void edge_pass2(
    const float* __restrict__ verts,
    const int*   __restrict__ esrc,
    const int*   __restrict__ edst,
    const float* __restrict__ row_sum,
    const float* __restrict__ col_sum,
    float* __restrict__ out,
    int E) {
  const int i = blockIdx.x * TPB + threadIdx.x;
  const int e = i * 4;
  if (e >= E) return;

  if (((E & 3) == 0) && (e + 3 < E)) {
    v4i s4 = *(const v4i*)(esrc + e);
    v4i d4 = *(const v4i*)(edst + e);

    float w0 = edge_weight(verts, s4.x, d4.x);
    float w1 = edge_weight(verts, s4.y, d4.y);
    float w2 = edge_weight(verts, s4.z, d4.z);
    float w3 = edge_weight(verts, s4.w, d4.w);

    // (0.5f*w) is exact (exponent decrement), so (0.5f*w)/sum is bit-identical
    // to the reference's 0.5f*(w/sum).
    v4f vr, vc;
    vr.x = (0.5f * w0) / row_sum[s4.x];
    vr.y = (0.5f * w1) / row_sum[s4.y];
    vr.z = (0.5f * w2) / row_sum[s4.z];
    vr.w = (0.5f * w3) / row_sum[s4.w];
    vc.x = (0.5f * w0) / col_sum[d4.x];
    vc.y = (0.5f * w1) / col_sum[d4.y];
    vc.z = (0.5f * w2) / col_sum[d4.z];
    vc.w = (0.5f * w3) / col_sum[d4.w];

    __builtin_nontemporal_store(vr, (v4f*)(out + e));       // e % 4 == 0
    __builtin_nontemporal_store(vc, (v4f*)(out + E + e));   // E % 4 == 0
  } else {
    int lim = e + 4 < E ? e + 4 : E;
    for (int k = e; k < lim; ++k) {
      int s = esrc[k], d = edst[k];
      float w = edge_weight(verts, s, d);
      __builtin_nontemporal_store((0.5f * w) / row_sum[s], out + k);
      __builtin_nontemporal_store((0.5f * w) / col_sum[d], out + E + k);
    }
  }
}

extern "C" void kernel_launch(void* const* d_in, const int* in_sizes, int n_in,
                              void* d_out, int out_size, void* d_ws, size_t ws_size,
                              hipStream_t stream) {
  // setup_inputs order: features[unused], vertices, edges, faces[unused]
  const float* verts = (const float*)d_in[1];
  const int*   edges = (const int*)d_in[2];   // [2,E] flat: src[E] then dst[E]
  const int E = in_sizes[2] / 2;
  const int N = in_sizes[1] / 3;

  const int* esrc = edges;
  const int* edst = edges + E;

  float* row_sum = (float*)d_ws;        // N floats
  float* col_sum = row_sum + N;         // N floats  (total 4MB of d_ws)
  float* out     = (float*)d_out;       // 2E floats

  int nsum = 2 * N;
  zero_sums<<<(nsum + TPB - 1) / TPB, TPB, 0, stream>>>(row_sum, nsum);

  int nthreads = (E + 3) / 4;
  int eblocks  = (nthreads + TPB - 1) / TPB;
  edge_pass1<<<eblocks, TPB, 0, stream>>>(verts, esrc, edst, row_sum, col_sum, E);
  edge_pass2<<<eblocks, TPB, 0, stream>>>(verts, esrc, edst, row_sum, col_sum, out, E);
}